// Actor_87144886436443
// MI455X (gfx1250) — compile-verified
//
#include <hip/hip_runtime.h>
#include <hip/hip_bf16.h>
#include <math.h>

// ---------------------------------------------------------------------------
// Problem constants (structure of the reference network; B derived at launch)
//   B=8192, N=50, T=20, S=D=3, H=128
// After folding BN + first-layer + head weights, per (b,n,t) we need only:
//   q[j]  = st·Aqs[:,j] + dy·Aqd[:,j] + cq[j]        (j = 0..8)
//   dl    = st·Dbs + dy·Dbd + db                     (binary-head logit diff)
// The 70 composed scalars are produced once by a WMMA GEMM (16x16, K=256).
// ---------------------------------------------------------------------------

typedef __attribute__((ext_vector_type(2))) float v2f;
typedef __attribute__((ext_vector_type(8))) float v8f;

#define NN  50
#define TT  20
#define NPARAM 70

// ---------------------------------------------------------------------------
// Setup kernel: one wave32. Builds
//   L (16 x 256):  rows 0-2  = scale_s[s]*Ws[:,s]   in cols 0..127
//                  rows 3-5  = scale_d[s]*Wd[:,s]   in cols 128..255
//                  row  6    = [cs | cd]  (BN-shift + bias pushed through Ws/Wd)
//   R (256 x 16):  cols 0-8 = Wq^T, cols 9-10 = Wb^T, rest zero
// and computes D = L @ R with 64 chained V_WMMA_F32_16X16X4_F32 ops.
// Output: params[70] in d_ws.
// ---------------------------------------------------------------------------
__global__ void compose_kernel(
    const float* __restrict__ Ws, const float* __restrict__ bs,
    const float* __restrict__ Wd, const float* __restrict__ bd,
    const float* __restrict__ Wq, const float* __restrict__ bq,
    const float* __restrict__ Wb, const float* __restrict__ bb,
    const float* __restrict__ g_s, const float* __restrict__ beta_s,
    const float* __restrict__ m_s, const float* __restrict__ v_s,
    const float* __restrict__ g_d, const float* __restrict__ beta_d,
    const float* __restrict__ m_d, const float* __restrict__ v_d,
    float* __restrict__ params)
{
    __shared__ float L[16][256];
    __shared__ float R[256][16];
    __shared__ float Dout[16][16];

    const int tid = threadIdx.x;  // 32 threads, one wave, no divergence pre-WMMA

    for (int i = tid; i < 16 * 256; i += 32) (&L[0][0])[i] = 0.0f;
    for (int i = tid; i < 256 * 16; i += 32) (&R[0][0])[i] = 0.0f;
    __syncthreads();

    // BN eval-mode fold: BN(x) = x*scale + shift
    float sc_s[3], sh_s[3], sc_d[3], sh_d[3];
    for (int s = 0; s < 3; ++s) {
        sc_s[s] = g_s[s] / sqrtf(v_s[s] + 1e-5f);
        sh_s[s] = beta_s[s] - m_s[s] * sc_s[s];
        sc_d[s] = g_d[s] / sqrtf(v_d[s] + 1e-5f);
        sh_d[s] = beta_d[s] - m_d[s] * sc_d[s];
    }

    for (int h = tid; h < 128; h += 32) {          // 128 % 32 == 0 -> uniform
        float cs = bs[h], cd = bd[h];
        for (int s = 0; s < 3; ++s) {
            float ws = Ws[h * 3 + s];              // Ws is [H,S] row-major
            float wd = Wd[h * 3 + s];
            L[s][h]           = sc_s[s] * ws;
            L[3 + s][128 + h] = sc_d[s] * wd;
            cs += sh_s[s] * ws;
            cd += sh_d[s] * wd;
        }
        L[6][h]       = cs;
        L[6][128 + h] = cd;
    }
    for (int k = tid; k < 256; k += 32) {          // 256 % 32 == 0 -> uniform
        for (int j = 0; j < 9; ++j) R[k][j] = Wq[j * 256 + k];  // Wq [9,256]
        R[k][9]  = Wb[k];            // Wb [2,256]
        R[k][10] = Wb[256 + k];
    }
    __syncthreads();

    // V_WMMA_F32_16X16X4_F32 operand layout (wave32):
    //   A (16x4):  lane m = tid&15; VGPR0/1 hold K = 2*(tid>>4) + {0,1}
    //   B (4x16):  lane n = tid&15; symmetric K split (assumed, mirrors A)
    //   C/D (16x16): VGPR r -> M = r + 8*(tid>>4), N = tid&15
    const int m  = tid & 15;
    const int kh = tid >> 4;      // 0 or 1
    v8f acc = {0.f, 0.f, 0.f, 0.f, 0.f, 0.f, 0.f, 0.f};
    for (int k0 = 0; k0 < 256; k0 += 4) {
        v2f a, b;
        a.x = L[m][k0 + 2 * kh];
        a.y = L[m][k0 + 2 * kh + 1];
        b.x = R[k0 + 2 * kh][m];
        b.y = R[k0 + 2 * kh + 1][m];
        acc = __builtin_amdgcn_wmma_f32_16x16x4_f32(
            false, a, false, b, (short)0, acc, false, false);
    }
    for (int r = 0; r < 8; ++r) Dout[r + 8 * kh][m] = acc[r];
    __syncthreads();

    // params layout:
    //  [ 0..26]  Aqs[s][j] = D[s][j]
    //  [27..53]  Aqd[s][j] = D[3+s][j]
    //  [54..62]  cq[j]     = D[6][j] + bq[j]
    //  [63..65]  Dbs[s]    = D[s][10]   - D[s][9]
    //  [66..68]  Dbd[s]    = D[3+s][10] - D[3+s][9]
    //  [69]      db        = D[6][10]-D[6][9] + bb[1]-bb[0]
    for (int i = tid; i < NPARAM; i += 32) {
        float val;
        if (i < 27)      { int s = i / 9,        j = i % 9;        val = Dout[s][j]; }
        else if (i < 54) { int s = (i - 27) / 9, j = (i - 27) % 9; val = Dout[3 + s][j]; }
        else if (i < 63) { int j = i - 54;  val = Dout[6][j] + bq[j]; }
        else if (i < 66) { int s = i - 63;  val = Dout[s][10] - Dout[s][9]; }
        else if (i < 69) { int s = i - 66;  val = Dout[3 + s][10] - Dout[3 + s][9]; }
        else             { val = Dout[6][10] - Dout[6][9] + bb[1] - bb[0]; }
        params[i] = val;
    }
}

// ---------------------------------------------------------------------------
// Block reductions: wave32 shuffle tree + 2-wave LDS combine (blockDim = 64).
// ---------------------------------------------------------------------------
__device__ __forceinline__ float blockReduceMax(float v, float* red, int tid) {
    for (int off = 16; off; off >>= 1) v = fmaxf(v, __shfl_xor(v, off, 32));
    __syncthreads();                      // protect red[] from prior readers
    if ((tid & 31) == 0) red[tid >> 5] = v;
    __syncthreads();
    return fmaxf(red[0], red[1]);
}
__device__ __forceinline__ float blockReduceSum(float v, float* red, int tid) {
    for (int off = 16; off; off >>= 1) v += __shfl_xor(v, off, 32);
    __syncthreads();
    if ((tid & 31) == 0) red[tid >> 5] = v;
    __syncthreads();
    return red[0] + red[1];
}

// ---------------------------------------------------------------------------
// Main kernel: one block per batch element b; lane = n (0..49 active of 64).
// Sequential T=20 scan; everything HBM-streaming (~500 MB total @ 23.3 TB/s).
// ---------------------------------------------------------------------------
__global__ void __launch_bounds__(64)
actor_kernel(const float* __restrict__ st,     // [B,N,3,T]
             const float* __restrict__ dyn,    // [B,N,3,T+1] (only t=0 used)
             const float* __restrict__ params, // [70] composed
             float* __restrict__ out,          // action ++ action_logp
             int B)
{
    __shared__ float P[NPARAM];
    __shared__ float red[2];

    const int tid = threadIdx.x;
    const int b   = blockIdx.x;

    for (int i = tid; i < NPARAM; i += 64) P[i] = params[i];

    // Warm L2/WGP$ with this block's 12 KB static tile (global_prefetch_b8).
    const float* sb = st + (size_t)b * NN * 3 * TT;
    for (int off = tid * 32; off < NN * 3 * TT; off += 64 * 32)
        __builtin_prefetch(sb + off, 0, 3);
    __syncthreads();

    const bool valid = tid < NN;
    const int  n     = valid ? tid : (NN - 1);   // clamp for safe dummy loads

    // dynamic[b,n,d,0]  at ((b*N+n)*3 + d)*(T+1)
    const float* dptr = dyn + (size_t)(b * NN + n) * 3 * (TT + 1);
    float dy0 = dptr[0], dy1 = dptr[TT + 1], dy2 = dptr[2 * (TT + 1)];

    const float* sp = sb + (size_t)n * 3 * TT;   // static[b,n,s,t] = sp[s*T+t]

    const size_t obase = (size_t)(b * NN + n) * 3 * TT;
    float* act = out + obase;
    float* alp = out + (size_t)B * NN * 3 * TT + obase;
    const float LOG_HALF = -0.69314718055994531f;

    for (int t = 0; t < TT; ++t) {
        const float st0 = sp[t], st1 = sp[TT + t], st2 = sp[2 * TT + t];

        // ---- 9-way head: logits, first-occurrence argmax, log(max softmax)
        float q[9], qmax = -INFINITY;
        int   qarg = 0;
        #pragma unroll
        for (int j = 0; j < 9; ++j) {
            float qq = P[54 + j]
                     + st0 * P[j]      + st1 * P[9 + j]  + st2 * P[18 + j]
                     + dy0 * P[27 + j] + dy1 * P[36 + j] + dy2 * P[45 + j];
            q[j] = qq;
            if (qq > qmax) { qmax = qq; qarg = j; }
        }
        float se = 0.0f;
        #pragma unroll
        for (int j = 0; j < 9; ++j) se += expf(q[j] - qmax);
        const float log_q = -logf(se);            // log(max softmax prob)
        const float ptrqf = (float)(qarg + 8);

        // ---- binary head -> sigmoid of logit diff, then softmax over N
        const float dl = P[69]
                       + st0 * P[63] + st1 * P[64] + st2 * P[65]
                       + dy0 * P[66] + dy1 * P[67] + dy2 * P[68];
        const float z    = 1.0f / (1.0f + expf(-dl));
        const float zmax = blockReduceMax(valid ? z : -INFINITY, red, tid);
        const float ez   = valid ? expf(z - zmax) : 0.0f;
        const float zsum = blockReduceSum(ez, red, tid);
        const float bdw  = ez / zsum;

        // ---- physics update (uses pre-update dyn)
        const float d2p  = dy1 + 0.05f * st2;
        const float d2sq = (d2p - 500.0f) * (d2p - 500.0f) + 100.0f;
        const float d2   = sqrtf(d2sq);
        const float rate = bdw * 10.0f * log2f(1.0f + 1e7f * st0 / d2sq);
        const float cand = 0.05f + ptrqf / (32.0f * rate);
        const float d0   = blockReduceMax(valid ? cand : -INFINITY, red, tid);
        const float d1   = d0 * st2 + dy1;

        if (valid) {
            act[t]            = 0.0f;       // ptr_s
            act[TT + t]       = ptrqf;      // ptr_q
            act[2 * TT + t]   = bdw;
            alp[t]            = LOG_HALF;   // logp_s
            alp[TT + t]       = log_q;
            alp[2 * TT + t]   = bdw;
        }
        dy0 = d0; dy1 = d1; dy2 = d2;
    }
}

// ---------------------------------------------------------------------------
extern "C" void kernel_launch(void* const* d_in, const int* in_sizes, int n_in,
                              void* d_out, int out_size, void* d_ws, size_t ws_size,
                              hipStream_t stream) {
    const float* st   = (const float*)d_in[0];   // static  [B,N,3,T]
    const float* dyn  = (const float*)d_in[1];   // dynamic [B,N,3,T+1]
    const float* Ws   = (const float*)d_in[2];
    const float* bs   = (const float*)d_in[3];
    const float* Wd   = (const float*)d_in[4];
    const float* bd   = (const float*)d_in[5];
    const float* Wq   = (const float*)d_in[6];
    const float* bq   = (const float*)d_in[7];
    const float* Wb   = (const float*)d_in[8];
    const float* bb   = (const float*)d_in[9];
    const float* g_s  = (const float*)d_in[10];
    const float* be_s = (const float*)d_in[11];
    const float* m_s  = (const float*)d_in[12];
    const float* v_s  = (const float*)d_in[13];
    const float* g_d  = (const float*)d_in[14];
    const float* be_d = (const float*)d_in[15];
    const float* m_d  = (const float*)d_in[16];
    const float* v_d  = (const float*)d_in[17];

    float* params = (float*)d_ws;
    float* outp   = (float*)d_out;

    const int B = in_sizes[0] / (NN * 3 * TT);   // 8192

    compose_kernel<<<1, 32, 0, stream>>>(Ws, bs, Wd, bd, Wq, bq, Wb, bb,
                                         g_s, be_s, m_s, v_s,
                                         g_d, be_d, m_d, v_d, params);
    actor_kernel<<<B, 64, 0, stream>>>(st, dyn, params, outp, B);
}